// RNN_12824772346400
// MI455X (gfx1250) — compile-verified
//
#include <hip/hip_runtime.h>
#include <math.h>

// Problem sizes (fixed by the reference).
#define NB 128   // batch
#define TT 512   // timesteps
#define DD 512   // input dim
#define HH 512   // hidden dim

typedef __attribute__((ext_vector_type(2))) float v2f;
typedef __attribute__((ext_vector_type(8))) float v8f;

// ---------------------------------------------------------------------------
// Shared pattern: each wave computes a 16x32 C tile (two 16x16 WMMA tiles,
// two independent f32 WMMA accumulation chains). B panels live in LDS as
// packed K-pairs, one 16-column panel per array so the ds_load_b64 pattern
// stays bank-conflict-free (lane half K offset = +128B = +32 banks).
// Fragment layouts per CDNA5 ISA:
//   A (16x4 f32): lane holds {A[m, k+2h], A[m, k+2h+1]}, m=lane&15, h=lane>>4
//   B (4x16):     lane holds {B[k+2h, n], B[k+2h+1, n]},  n=lane&15
//   C (16x16):    elem i -> row = i + 8h, col = lane&15
// ---------------------------------------------------------------------------

// Cooperative fill of one 16-column B panel: Bs[k2*16+n] = {B[2k2,c], B[2k2+1,c]}
__device__ inline void fill_bpanel(v2f* Bs, const float* __restrict__ B,
                                   int K, int colbase) {
    for (int i = threadIdx.x; i < (K / 2) * 16; i += 256) {
        const int k2 = i >> 4;
        const int n  = i & 15;
        v2f p;
        p.x = B[(size_t)(2 * k2) * HH + colbase + n];
        p.y = B[(size_t)(2 * k2 + 1) * HH + colbase + n];
        Bs[i] = p;
    }
}

// ---------------------------------------------------------------------------
// Kernel 1: out[m, :] = x[m, :] @ Wx + b    (M=65536, K=512, N=512)
// Block = 8 waves = 8 M tiles x one 32-column N panel. Grid = 512 x 16.
// ---------------------------------------------------------------------------
__global__ __launch_bounds__(256) void proj_kernel(const float* __restrict__ x,
                                                   const float* __restrict__ Wx,
                                                   const float* __restrict__ bias,
                                                   float* __restrict__ out) {
    __shared__ v2f Bs0[(DD / 2) * 16];   // 32 KB: Wx[:, nbase..nbase+15]
    __shared__ v2f Bs1[(DD / 2) * 16];   // 32 KB: Wx[:, nbase+16..nbase+31]

    const int npair = blockIdx.x & 15;   // 16 column pairs (32 cols each)
    const int mblk  = blockIdx.x >> 4;   // 512 blocks of 128 rows
    const int nbase = npair * 32;

    fill_bpanel(Bs0, Wx, DD, nbase);
    fill_bpanel(Bs1, Wx, DD, nbase + 16);
    __syncthreads();

    const int wave  = threadIdx.x >> 5;
    const int lane  = threadIdx.x & 31;
    const int mcol  = lane & 15;
    const int hsel  = lane >> 4;
    const int mbase = mblk * 128 + wave * 16;

    const float* arow = x + (size_t)(mbase + mcol) * DD;

    // Warm the output lines this wave will write (1 x 128B line per C row).
    if (lane < 16) {
        __builtin_prefetch(out + (size_t)(mbase + lane) * HH + nbase, 1, 1);
    }

    v8f acc0 = {};
    v8f acc1 = {};
#pragma unroll 4
    for (int k = 0; k < DD; k += 4) {
        const int k2i = (k / 2 + hsel) * 16 + mcol;
        v2f a  = *(const v2f*)(arow + k + 2 * hsel);
        v2f b0 = Bs0[k2i];
        v2f b1 = Bs1[k2i];
        acc0 = __builtin_amdgcn_wmma_f32_16x16x4_f32(
            false, a, false, b0, (short)0, acc0, false, false);
        acc1 = __builtin_amdgcn_wmma_f32_16x16x4_f32(
            false, a, false, b1, (short)0, acc1, false, false);
    }

    const float bv0 = bias[nbase + mcol];
    const float bv1 = bias[nbase + 16 + mcol];
#pragma unroll
    for (int i = 0; i < 8; ++i) {
        const size_t ro = (size_t)(mbase + i + 8 * hsel) * HH + nbase + mcol;
        out[ro]      = acc0[i] + bv0;
        out[ro + 16] = acc1[i] + bv1;
    }
}

// ---------------------------------------------------------------------------
// Kernel 2: hbuf0 = h0, barrier counter = 0.
// ---------------------------------------------------------------------------
__global__ void init_kernel(const float* __restrict__ h0,
                            float* __restrict__ hbuf0,
                            unsigned* __restrict__ cnt) {
    const int i = blockIdx.x * blockDim.x + threadIdx.x;
    if (i < NB * HH) hbuf0[i] = h0[i];
    if (i == 0) *cnt = 0u;
}

// ---------------------------------------------------------------------------
// Kernel 3: persistent scan. 16 workgroups, one 32-column panel of H each.
// Wh panels staged in LDS once and reused for all 512 steps. 8 waves cover
// the 8 M tiles of the 128-row batch; each wave owns a 16x32 C tile with two
// independent WMMA chains. Grid-wide step barrier via monotone counter.
// out updated in place: slice t holds xWx+b on entry, h_t on exit.
// ---------------------------------------------------------------------------
__global__ __launch_bounds__(256) void rnn_scan_kernel(const float* __restrict__ Wh,
                                                       float* __restrict__ out,
                                                       float* __restrict__ hbuf,
                                                       unsigned* __restrict__ cnt) {
    __shared__ v2f Bs0[(HH / 2) * 16];   // 32 KB
    __shared__ v2f Bs1[(HH / 2) * 16];   // 32 KB

    const int nbase = blockIdx.x * 32;   // 16 blocks x 32 cols = 512

    fill_bpanel(Bs0, Wh, HH, nbase);
    fill_bpanel(Bs1, Wh, HH, nbase + 16);
    __syncthreads();

    const int wave  = threadIdx.x >> 5;
    const int lane  = threadIdx.x & 31;
    const int mcol  = lane & 15;
    const int hsel  = lane >> 4;
    const int mbase = wave * 16;         // 8 waves * 16 = 128 batch rows

    float* buf0 = hbuf;
    float* buf1 = hbuf + NB * HH;

    for (int t = 0; t < TT; ++t) {
        const float* hprev = (t & 1) ? buf1 : buf0;
        float*       hnext = (t & 1) ? buf0 : buf1;

        // Warm this step's xw lines (one 128B line per C row of this wave).
        if (lane < 16) {
            __builtin_prefetch(
                out + ((size_t)(mbase + lane) * TT + t) * HH + nbase, 1, 1);
        }

        const float* arow = hprev + (size_t)(mbase + mcol) * HH;
        v8f acc0 = {};
        v8f acc1 = {};
#pragma unroll 4
        for (int k = 0; k < HH; k += 4) {
            const int k2i = (k / 2 + hsel) * 16 + mcol;
            v2f a  = *(const v2f*)(arow + k + 2 * hsel);
            v2f b0 = Bs0[k2i];
            v2f b1 = Bs1[k2i];
            acc0 = __builtin_amdgcn_wmma_f32_16x16x4_f32(
                false, a, false, b0, (short)0, acc0, false, false);
            acc1 = __builtin_amdgcn_wmma_f32_16x16x4_f32(
                false, a, false, b1, (short)0, acc1, false, false);
        }

#pragma unroll
        for (int i = 0; i < 8; ++i) {
            const int    row = mbase + i + 8 * hsel;                // batch idx
            const size_t oi  = ((size_t)row * TT + t) * HH + nbase + mcol;
            const float  v0  = tanhf(acc0[i] + out[oi]);
            const float  v1  = tanhf(acc1[i] + out[oi + 16]);
            out[oi]      = v0;
            out[oi + 16] = v1;
            hnext[(size_t)row * HH + nbase + mcol]      = v0;
            hnext[(size_t)row * HH + nbase + 16 + mcol] = v1;
        }

        // ---- grid-wide step barrier (release/acquire, agent scope) ----
        __threadfence();
        __syncthreads();
        if (threadIdx.x == 0) {
            __hip_atomic_fetch_add(cnt, 1u, __ATOMIC_ACQ_REL,
                                   __HIP_MEMORY_SCOPE_AGENT);
            const unsigned target = 16u * (unsigned)(t + 1);
            while (__hip_atomic_load(cnt, __ATOMIC_ACQUIRE,
                                     __HIP_MEMORY_SCOPE_AGENT) < target) {
                __builtin_amdgcn_s_sleep(2);
            }
        }
        __syncthreads();
        __threadfence();
    }
}

// ---------------------------------------------------------------------------
extern "C" void kernel_launch(void* const* d_in, const int* in_sizes, int n_in,
                              void* d_out, int out_size, void* d_ws, size_t ws_size,
                              hipStream_t stream) {
    const float* x  = (const float*)d_in[0];   // (N,T,D)
    const float* h0 = (const float*)d_in[1];   // (N,H)
    const float* Wx = (const float*)d_in[2];   // (D,H)
    const float* Wh = (const float*)d_in[3];   // (H,H)
    const float* b  = (const float*)d_in[4];   // (H,)

    float*    out  = (float*)d_out;            // (N,T,H)
    float*    hbuf = (float*)d_ws;             // 2 * N*H floats (double buffer)
    unsigned* cnt  = (unsigned*)((char*)d_ws + (size_t)2 * NB * HH * sizeof(float));

    // 1) out = x @ Wx + b : 512 M-blocks x 16 column-pair panels
    hipLaunchKernelGGL(proj_kernel, dim3((NB * TT / 128) * 16), dim3(256), 0, stream,
                       x, Wx, b, out);
    // 2) hbuf[0] = h0, counter = 0
    hipLaunchKernelGGL(init_kernel, dim3((NB * HH + 255) / 256), dim3(256), 0, stream,
                       h0, hbuf, cnt);
    // 3) persistent sequential scan over T with grid barrier per step
    hipLaunchKernelGGL(rnn_scan_kernel, dim3(16), dim3(256), 0, stream,
                       Wh, out, hbuf, cnt);
}